// SSDLoss_55267639165205
// MI455X (gfx1250) — compile-verified
//
#include <hip/hip_runtime.h>
#include <hip/hip_bf16.h>
#include <stdint.h>

#define BB   8
#define AA   65536
#define GG   32
#define CCL  21
#define BGC  4
#define MAXP 16384     // cap on stored matched pairs (realistic N ~ tens)
#define RECS 256       // records staged per block in k_confneg
#define NEGB 16        // gridDim.x of k_negsum (compile-time for uniform WMMA loop)

typedef __attribute__((ext_vector_type(2))) float v2f;
typedef __attribute__((ext_vector_type(8))) float v8f;
typedef int v4i __attribute__((vector_size(16)));   // matches builtin param type

#if defined(__HIP_DEVICE_COMPILE__)
#if __has_builtin(__builtin_amdgcn_wmma_f32_16x16x4_f32)
#define HAVE_WMMA_F32X4 1
#endif
#if __has_builtin(__builtin_amdgcn_global_load_async_to_lds_b128)
#define HAVE_ASYNC_LDS 1
#endif
#if __has_builtin(__builtin_amdgcn_s_wait_asynccnt)
#define HAVE_WAIT_ASYNC 1
#endif
#endif

// ---------------- init ----------------
__global__ void k_init(int* counters, float* sums) {
    int t = threadIdx.x;
    if (t < 8) counters[t] = 0;
    if (t < 3 * BB) sums[t] = 0.0f;
}

// ---------------- IoU match: pair list + is_neg + N + neg_num ----------------
__global__ void k_match(const float* __restrict__ dflt, const float* __restrict__ gtb,
                        int* counters, int* pA, int* pG, int* pF,
                        unsigned char* isNeg) {
    __shared__ float sg[GG * 4];
    int t = threadIdx.x;
    if (t < GG * 4) sg[t] = gtb[t];
    __syncthreads();
    int a = blockIdx.x * blockDim.x + t;
    float dcx = dflt[a * 4 + 0], dcy = dflt[a * 4 + 1];
    float dw  = dflt[a * 4 + 2], dh  = dflt[a * 4 + 3];
    float dx1 = dcx - 0.5f * dw, dx2 = dcx + 0.5f * dw;
    float dy1 = dcy - 0.5f * dh, dy2 = dcy + 0.5f * dh;
    float areaD = dw * dh;
    int any = 0;
    for (int g = 0; g < GG; ++g) {
        float gcx = sg[g * 4 + 0], gcy = sg[g * 4 + 1];
        float gw  = sg[g * 4 + 2], gh  = sg[g * 4 + 3];
        float w = fminf(dx2, gcx + 0.5f * gw) - fmaxf(dx1, gcx - 0.5f * gw);
        float h = fminf(dy2, gcy + 0.5f * gh) - fmaxf(dy1, gcy - 0.5f * gh);
        float inter = (w > 0.0f && h > 0.0f) ? w * h : 0.0f;
        float iou = inter / (areaD + gw * gh - inter);
        if (iou > 0.5f) {
            any = 1;
            int idx = atomicAdd(&counters[0], 1);     // N
            if (idx < MAXP) { pA[idx] = a; pG[idx] = g; pF[idx] = a * GG + g; }
        }
    }
    isNeg[a] = (unsigned char)(!any);
    if (!any) atomicAdd(&counters[1], 1);             // neg_num
}

// ---------------- pn / nn ----------------
__global__ void k_scalars(int* counters) {
    if (threadIdx.x == 0 && blockIdx.x == 0) {
        long long N = counters[0], neg = counters[1];
        int pn, nn;
        if (3 * N > neg) { pn = (int)(neg / 3); nn = (int)neg; }
        else             { pn = (int)N;         nn = (int)(3 * N); }
        counters[3] = pn; counters[4] = nn;
    }
}

// ---------------- conf_neg keys: 52MB stream, staged via async global->LDS ----------------
__global__ void k_confneg(const float* __restrict__ out, const unsigned char* __restrict__ isNeg,
                          unsigned* __restrict__ keys) {
    int t = threadIdx.x;
    unsigned rec0 = blockIdx.x * RECS;           // 256 contiguous (b,a) records
#ifdef HAVE_ASYNC_LDS
    __shared__ float tile[RECS * 25];            // 25600 B, 16B-aligned chunks
    {
        const float* gbase = out + (size_t)rec0 * 25;
        // 1600 x b128 chunks, coalesced, ASYNCcnt-tracked, no VGPR staging
        for (int i = t; i < (RECS * 25) / 4; i += blockDim.x) {
            __builtin_amdgcn_global_load_async_to_lds_b128(
                (__attribute__((address_space(1))) v4i*)(gbase + (size_t)i * 4),
                (__attribute__((address_space(3))) v4i*)(tile + (size_t)i * 4),
                0, 0);
        }
#ifdef HAVE_WAIT_ASYNC
        __builtin_amdgcn_s_wait_asynccnt(0);
#else
        asm volatile("s_wait_asynccnt 0x0" ::: "memory");
#endif
        __syncthreads();
    }
    const float* p = tile + t * 25 + 4;          // stride 25 (odd) -> bank-conflict-free
#else
    const float* p = out + (size_t)(rec0 + t) * 25 + 4;
    __builtin_prefetch(p + 25 * 1024, 0, 0);
#endif
    float l[CCL];
    float m = -1e30f;
#pragma unroll
    for (int c = 0; c < CCL; ++c) { l[c] = p[c]; m = fmaxf(m, l[c]); }
    float s = 0.0f;
#pragma unroll
    for (int c = 0; c < CCL; ++c) s += expf(l[c] - m);
    float conf = (m + logf(s)) - l[BGC];         // -logp[BG]  (>= 0)
    unsigned gid = rec0 + t;
    keys[gid] = isNeg[gid & (AA - 1)] ? __float_as_uint(conf) : 0u;
}

// ---------------- per (image, pair): conf_pos and smooth-L1 ----------------
__global__ void k_pospairs(const float* __restrict__ out, const float* __restrict__ dflt,
                           const float* __restrict__ gtb, const int* __restrict__ gtc,
                           const int* counters, const int* pA, const int* pG,
                           float* posConf, float* posSl1) {
    int b = blockIdx.y;
    int j = blockIdx.x * blockDim.x + threadIdx.x;
    int N = counters[0]; if (N > MAXP) N = MAXP;
    if (j >= N) return;
    int a = pA[j], g = pG[j];
    float dcx = dflt[a * 4], dcy = dflt[a * 4 + 1], dw = dflt[a * 4 + 2], dh = dflt[a * 4 + 3];
    float gcx = gtb[g * 4],  gcy = gtb[g * 4 + 1],  gw = gtb[g * 4 + 2],  gh = gtb[g * 4 + 3];
    float de0 = (gcx - dcx) / dw, de1 = (gcy - dcy) / dh;
    float de2 = logf(gw / dw),    de3 = logf(gh / dh);
    const float* pb = out + ((size_t)b * AA + a) * 25;
    float d0 = pb[0] - de0, d1 = pb[1] - de1, d2 = pb[2] - de2, d3 = pb[3] - de3;
    float s1 = 0.0f, ad;
    ad = fabsf(d0); s1 += (ad < 1.0f) ? 0.5f * d0 * d0 : ad - 0.5f;
    ad = fabsf(d1); s1 += (ad < 1.0f) ? 0.5f * d1 * d1 : ad - 0.5f;
    ad = fabsf(d2); s1 += (ad < 1.0f) ? 0.5f * d2 * d2 : ad - 0.5f;
    ad = fabsf(d3); s1 += (ad < 1.0f) ? 0.5f * d3 * d3 : ad - 0.5f;
    float m = -1e30f, l[CCL];
#pragma unroll
    for (int c = 0; c < CCL; ++c) { l[c] = pb[4 + c]; m = fmaxf(m, l[c]); }
    float se = 0.0f;
#pragma unroll
    for (int c = 0; c < CCL; ++c) se += expf(l[c] - m);
    float lse = m + logf(se);
    int cls = gtc[g];
    posConf[(size_t)b * MAXP + j] = lse - pb[4 + cls];
    posSl1[(size_t)b * MAXP + j]  = s1;
}

// ---------------- exact k-th-largest via 4-pass radix select (1 block / image) ----------------
// mode 0: negatives (stride AA, L=AA, k=nn); mode 1: positives (stride MAXP, L=N, k=pn)
__global__ void k_select(const unsigned* __restrict__ keysBase, const int* counters,
                         int mode, unsigned* sel) {
    int b = blockIdx.x;
    int L, k;
    if (mode == 0) { L = AA; k = counters[4]; }
    else { L = counters[0]; if (L > MAXP) L = MAXP; k = counters[3]; if (k > L) k = L; }
    const unsigned* keys = keysBase + (size_t)b * (mode == 0 ? AA : MAXP);
    int t = threadIdx.x;
    if (k <= 0) { if (t == 0) { sel[2 * b] = 0xFFFFFFFFu; sel[2 * b + 1] = 0u; } return; }
    __shared__ unsigned hist[256];
    __shared__ unsigned s_prefix;
    __shared__ int s_krem;
    if (t == 0) { s_prefix = 0u; s_krem = k; }
    for (int p = 0; p < 4; ++p) {
        int shift = 24 - 8 * p;
        for (int i = t; i < 256; i += blockDim.x) hist[i] = 0u;
        __syncthreads();
        unsigned pref = s_prefix;
        for (int i = t; i < L; i += blockDim.x) {
            unsigned key = keys[i];
            bool part = (p == 0) || ((key >> (shift + 8)) == (pref >> (shift + 8)));
            if (part) atomicAdd(&hist[(key >> shift) & 255u], 1u);
        }
        __syncthreads();
        if (t == 0) {
            unsigned cum = 0; int selb = 0; unsigned krem = (unsigned)s_krem;
            for (int bkt = 255; bkt >= 0; --bkt) {
                unsigned c = hist[bkt];
                if (cum + c >= krem) { selb = bkt; break; }
                cum += c;
            }
            s_krem = (int)(krem - cum);
            s_prefix = pref | ((unsigned)selb << shift);
        }
        __syncthreads();
    }
    if (t == 0) { sel[2 * b] = s_prefix; sel[2 * b + 1] = (unsigned)s_krem; }
}

// ---------------- positive sums (conf + sl1), stable tie order by flat index ----------------
__global__ void k_possum(const int* counters, const float* posConf, const float* posSl1,
                         const int* pF, const unsigned* sel, float* sums) {
    int b = blockIdx.y;
    int j = blockIdx.x * blockDim.x + threadIdx.x;
    int N = counters[0]; if (N > MAXP) N = MAXP;
    if (j >= N) return;
    unsigned T = sel[2 * b], need = sel[2 * b + 1];
    float conf = posConf[(size_t)b * MAXP + j];
    unsigned key = __float_as_uint(conf);
    bool take = false;
    if (key > T) take = true;
    else if (key == T && need > 0) {
        int myf = pF[j], rank = 0;
        for (int j2 = 0; j2 < N; ++j2)
            if (__float_as_uint(posConf[(size_t)b * MAXP + j2]) == T && pF[j2] < myf) rank++;
        take = (rank < (int)need);
    }
    if (take) {
        atomicAdd(&sums[b], conf);
        atomicAdd(&sums[BB + b], posSl1[(size_t)b * MAXP + j]);
    }
}

// ---------------- negative sum over threshold: WMMA f32 16x16x4 reduction engine ----------------
// Launched with gridDim.x == NEGB, blockDim.x == 256 (8 waves): compile-time trip count
// keeps wave control flow fully uniform (EXEC all-ones for every v_wmma issue).
__global__ void k_negsum(const unsigned* __restrict__ keys, const unsigned* __restrict__ sel,
                         float* sums) {
    int b = blockIdx.y;
    unsigned T = sel[2 * b];
    const unsigned* kb = keys + (size_t)b * AA;
    int lane = threadIdx.x & 31;
    int wave = threadIdx.x >> 5;
    int gw = blockIdx.x * 8 + wave;          // 128 waves total
    const int STEP = NEGB * 8 * 64;          // 8192
    float s;
#ifdef HAVE_WMMA_F32X4
    v8f c = {0.f, 0.f, 0.f, 0.f, 0.f, 0.f, 0.f, 0.f};
    v2f onesv = {1.0f, 1.0f};
#pragma unroll
    for (int it = 0; it < AA / STEP; ++it) { // 8 iterations, structurally uniform
        int i = gw * 64 + it * STEP;
        unsigned k0 = kb[i + lane * 2], k1 = kb[i + lane * 2 + 1];
        v2f a;
        a.x = (k0 > T) ? __uint_as_float(k0) : 0.0f;
        a.y = (k1 > T) ? __uint_as_float(k1) : 0.0f;
        // D = A(16x4) * ones(4x16) + C : every column carries the row-sums
        c = __builtin_amdgcn_wmma_f32_16x16x4_f32(false, a, false, onesv, (short)0, c,
                                                  false, false);
    }
    s = (c[0] + c[1] + c[2] + c[3] + c[4] + c[5] + c[6] + c[7]) * 0.0625f; // /16 replication
#else
    s = 0.0f;
    for (int it = 0; it < AA / STEP; ++it) {
        int i = gw * 64 + it * STEP;
        unsigned k0 = kb[i + lane * 2], k1 = kb[i + lane * 2 + 1];
        if (k0 > T) s += __uint_as_float(k0);
        if (k1 > T) s += __uint_as_float(k1);
    }
#endif
    __shared__ float red[256];
    red[threadIdx.x] = s;
    __syncthreads();
    for (int off = blockDim.x >> 1; off > 0; off >>= 1) {
        if (threadIdx.x < (unsigned)off) red[threadIdx.x] += red[threadIdx.x + off];
        __syncthreads();
    }
    if (threadIdx.x == 0) atomicAdd(&sums[2 * BB + b], red[0]);
}

// ---------------- finalize: add tie mass for negatives, mean over images ----------------
__global__ void k_final(const int* counters, const float* sums, const unsigned* selNeg,
                        float* out) {
    if (threadIdx.x != 0 || blockIdx.x != 0) return;
    float Nf = (float)counters[0];
    float acc = 0.0f;
    for (int b = 0; b < BB; ++b) {
        float neg = sums[2 * BB + b];
        unsigned need = selNeg[2 * b + 1];
        if (need) neg += (float)need * __uint_as_float(selNeg[2 * b]); // ties all equal T
        acc += (sums[BB + b] + sums[b]) / Nf + neg / Nf;
    }
    out[0] = acc / (float)BB;
}

extern "C" void kernel_launch(void* const* d_in, const int* in_sizes, int n_in,
                              void* d_out, int out_size, void* d_ws, size_t ws_size,
                              hipStream_t stream) {
    const float* outputs = (const float*)d_in[0];   // (8, 65536, 25)
    const float* dflt    = (const float*)d_in[1];   // (65536, 4)
    const float* gtb     = (const float*)d_in[2];   // (32, 4)
    const int*   gtc     = (const int*)d_in[3];     // (32,)

    char* base = (char*)d_ws; size_t off = 0;
    auto take = [&](size_t bytes) -> char* {
        char* p = base + off; off = (off + bytes + 255) & ~(size_t)255; return p;
    };
    unsigned*      keysNeg = (unsigned*)take((size_t)BB * AA * 4);   // 2 MB
    int*           counters = (int*)take(64);
    int*           pA = (int*)take(MAXP * 4);
    int*           pG = (int*)take(MAXP * 4);
    int*           pF = (int*)take(MAXP * 4);
    unsigned char* isNeg = (unsigned char*)take(AA);
    float*         posConf = (float*)take((size_t)BB * MAXP * 4);
    float*         posSl1  = (float*)take((size_t)BB * MAXP * 4);
    unsigned*      selNeg = (unsigned*)take(2 * BB * 4);
    unsigned*      selPos = (unsigned*)take(2 * BB * 4);
    float*         sums = (float*)take(3 * BB * 4);

    k_init<<<1, 256, 0, stream>>>(counters, sums);
    k_match<<<AA / 256, 256, 0, stream>>>(dflt, gtb, counters, pA, pG, pF, isNeg);
    k_scalars<<<1, 32, 0, stream>>>(counters);
    k_confneg<<<(BB * AA) / RECS, RECS, 0, stream>>>(outputs, isNeg, keysNeg);
    dim3 gP(MAXP / 256, BB);
    k_pospairs<<<gP, 256, 0, stream>>>(outputs, dflt, gtb, gtc, counters, pA, pG, posConf, posSl1);
    k_select<<<BB, 256, 0, stream>>>(keysNeg, counters, 0, selNeg);
    k_select<<<BB, 256, 0, stream>>>((const unsigned*)posConf, counters, 1, selPos);
    k_possum<<<gP, 256, 0, stream>>>(counters, posConf, posSl1, pF, selPos, sums);
    dim3 gN(NEGB, BB);
    k_negsum<<<gN, 256, 0, stream>>>(keysNeg, selNeg, sums);
    k_final<<<1, 32, 0, stream>>>(counters, sums, selNeg, (float*)d_out);
}